// GNN_Backbone_60026462929462
// MI455X (gfx1250) — compile-verified
//
#include <hip/hip_runtime.h>

#define N_NODES 100000
#define N_EDGES 1600000
#define HID 64
#define BN_EPS 1e-5f

typedef __attribute__((ext_vector_type(2))) float v2f;
typedef __attribute__((ext_vector_type(8))) float v8f;

__device__ __forceinline__ void atomic_add_f32(float* p, float v) {
  __hip_atomic_fetch_add(p, v, __ATOMIC_RELAXED, __HIP_MEMORY_SCOPE_AGENT);
}

// ---------------- init: deg=1 (self loop), winner=-1 ----------------
__global__ void init_kernel(float* __restrict__ deg, int* __restrict__ winner) {
  int n = blockIdx.x * blockDim.x + threadIdx.x;
  if (n < N_NODES) { deg[n] = 1.0f; winner[n] = -1; }
}

// ---------------- per-edge: degree accum + last-write-wins winner ----------------
__global__ void edge_prep_kernel(const int* __restrict__ src, const int* __restrict__ dst,
                                 const float* __restrict__ ea,
                                 float* __restrict__ deg, int* __restrict__ winner) {
  int j = blockIdx.x * blockDim.x + threadIdx.x;
  if (j >= N_EDGES) return;
  atomic_add_f32(&deg[dst[j]], ea[j]);
  atomicMax(&winner[src[j]], j);            // src pass: priority j
  atomicMax(&winner[dst[j]], j + N_EDGES);  // dst pass overrides src pass
}

__global__ void dinv_kernel(float* __restrict__ deg) {
  int n = blockIdx.x * blockDim.x + threadIdx.x;
  if (n < N_NODES) deg[n] = rsqrtf(deg[n]);  // deg >= 1 always
}

// ---------------- Out[M x 64] = A[M x 64] @ W[64 x 64] (+bias), f32 WMMA ----------------
// block = 128 threads = 4 waves; wave w computes the 16x16 tile at cols [16w,16w+16).
// grid.x = M/16 (M = 100000 = 6250*16).
__global__ void gemm_n64_kernel(const float* __restrict__ A, const float* __restrict__ W,
                                const float* __restrict__ bias, float* __restrict__ Out) {
  const int wave  = threadIdx.x >> 5;
  const int lane  = threadIdx.x & 31;
  const int r     = lane & 15;      // A row / B col / D col within tile
  const int khalf = lane >> 4;      // K-half select
  const int colg  = wave * 16 + r;  // global output column (0..63)
  const long rowBase = (long)(blockIdx.x * 16 + r) * HID;

  v8f c;
  float cinit = bias ? bias[colg] : 0.0f;
#pragma unroll
  for (int v = 0; v < 8; ++v) c[v] = cinit;

#pragma unroll
  for (int k0 = 0; k0 < HID; k0 += 4) {
    // A fragment: lane holds K = k0 + 2*khalf + {0,1} of row r  (contiguous -> 8B load)
    v2f a = *(const v2f*)(A + rowBase + k0 + 2 * khalf);
    // B fragment: lane holds rows K = k0 + 2*khalf + {0,1}, column colg
    v2f b;
    b.x = W[(k0 + 2 * khalf + 0) * HID + colg];
    b.y = W[(k0 + 2 * khalf + 1) * HID + colg];
    c = __builtin_amdgcn_wmma_f32_16x16x4_f32(false, a, false, b, (short)0, c, false, false);
  }

#pragma unroll
  for (int v = 0; v < 8; ++v) {
    int rowOut = v + 8 * khalf;  // D layout: VGPR v -> row v (+8 for upper half-wave)
    Out[(long)(blockIdx.x * 16 + rowOut) * HID + colg] = c[v];
  }
}

// ---------------- apply winner edge embedding + relu at winner nodes ----------------
__global__ void apply_edge_kernel(float* __restrict__ h, const int* __restrict__ winner,
                                  const float* __restrict__ ea,
                                  const float* __restrict__ ew, const float* __restrict__ eb) {
  long gid = (long)blockIdx.x * blockDim.x + threadIdx.x;
  if (gid >= (long)N_NODES * HID) return;
  int n = (int)(gid >> 6), c = (int)(gid & 63);
  int w = winner[n];
  if (w < 0) return;
  int j = (w >= N_EDGES) ? (w - N_EDGES) : w;
  float v = h[gid] + ea[j] * ew[c] + eb[c];
  h[gid] = fmaxf(v, 0.0f);
}

// ---------------- agg = conv_b + dinv[n]^2 * xw[n]  (self-loop term) ----------------
__global__ void agg_init_kernel(const float* __restrict__ xw, const float* __restrict__ dinv,
                                const float* __restrict__ cb, float* __restrict__ agg) {
  long gid = (long)blockIdx.x * blockDim.x + threadIdx.x;
  if (gid >= (long)N_NODES * HID) return;
  int n = (int)(gid >> 6), c = (int)(gid & 63);
  float di = dinv[n];
  agg[gid] = cb[c] + di * di * xw[gid];
}

// ---------------- agg[dst] += dinv[src]*ea*dinv[dst] * xw[src]  (per edge,channel) ----------------
__global__ void scatter_kernel(const float* __restrict__ xw, const float* __restrict__ dinv,
                               const int* __restrict__ src, const int* __restrict__ dst,
                               const float* __restrict__ ea, float* __restrict__ agg) {
  long gid = (long)blockIdx.x * blockDim.x + threadIdx.x;
  if (gid >= (long)N_EDGES * HID) return;
  int j = (int)(gid >> 6), c = (int)(gid & 63);
  int s = src[j], d = dst[j];
  float coef = dinv[s] * ea[j] * dinv[d];
  atomic_add_f32(&agg[(long)d * HID + c], coef * xw[(long)s * HID + c]);
}

__global__ void zero_sums_kernel(float* __restrict__ gsum, float* __restrict__ gsumsq) {
  if (threadIdx.x < HID) { gsum[threadIdx.x] = 0.0f; gsumsq[threadIdx.x] = 0.0f; }
}

// ---------------- per-channel sum / sumsq over nodes ----------------
// blockDim=256 -> each thread's channel (gid & 63) is invariant across grid-stride iters.
__global__ void bn_stats_kernel(const float* __restrict__ agg,
                                float* __restrict__ gsum, float* __restrict__ gsumsq) {
  __shared__ float s1[256];
  __shared__ float s2[256];
  const long total = (long)N_NODES * HID;
  const long stride = (long)gridDim.x * blockDim.x;
  float sum = 0.0f, sq = 0.0f;
  for (long gid = (long)blockIdx.x * blockDim.x + threadIdx.x; gid < total; gid += stride) {
    float v = agg[gid];
    sum += v;
    sq += v * v;
  }
  s1[threadIdx.x] = sum;
  s2[threadIdx.x] = sq;
  __syncthreads();
  if (threadIdx.x < HID) {
    float a = s1[threadIdx.x] + s1[threadIdx.x + 64] + s1[threadIdx.x + 128] + s1[threadIdx.x + 192];
    float b = s2[threadIdx.x] + s2[threadIdx.x + 64] + s2[threadIdx.x + 128] + s2[threadIdx.x + 192];
    atomic_add_f32(&gsum[threadIdx.x], a);
    atomic_add_f32(&gsumsq[threadIdx.x], b);
  }
}

// ---------------- BN + gamma/beta (+relu for layer 0) -> h and output slice ----------------
__global__ void bn_norm_kernel(const float* __restrict__ agg, const float* __restrict__ gsum,
                               const float* __restrict__ gsumsq,
                               const float* __restrict__ gamma, const float* __restrict__ beta,
                               float* __restrict__ h, float* __restrict__ out, int layer) {
  long gid = (long)blockIdx.x * blockDim.x + threadIdx.x;
  if (gid >= (long)N_NODES * HID) return;
  int n = (int)(gid >> 6), c = (int)(gid & 63);
  const float invN = 1.0f / (float)N_NODES;
  float mean = gsum[c] * invN;
  float var = gsumsq[c] * invN - mean * mean;  // biased variance
  float inv = rsqrtf(var + BN_EPS);
  float v = (agg[gid] - mean) * inv * gamma[c] + beta[c];
  if (layer == 0) {
    v = fmaxf(v, 0.0f);  // relu on all but last layer
    h[gid] = v;          // input to next layer
  }
  out[(long)n * (2 * HID) + layer * HID + c] = v;
}

extern "C" void kernel_launch(void* const* d_in, const int* in_sizes, int n_in,
                              void* d_out, int out_size, void* d_ws, size_t ws_size,
                              hipStream_t stream) {
  const float* x     = (const float*)d_in[0];
  const int*   ei    = (const int*)d_in[1];
  const float* ea    = (const float*)d_in[2];
  const float* linW  = (const float*)d_in[3];
  const float* linb  = (const float*)d_in[4];
  const float* ew    = (const float*)d_in[5];
  const float* eb    = (const float*)d_in[6];
  const float* convW = (const float*)d_in[7];
  const float* convb = (const float*)d_in[8];
  const float* gamma = (const float*)d_in[9];
  const float* beta  = (const float*)d_in[10];
  float* out = (float*)d_out;

  // workspace carve-up
  float* h      = (float*)d_ws;                    // [N, 64]
  float* xw     = h + (long)N_NODES * HID;         // [N, 64]
  float* agg    = xw + (long)N_NODES * HID;        // [N, 64]
  float* dinv   = agg + (long)N_NODES * HID;       // [N]  (holds deg, then 1/sqrt(deg))
  int*   winner = (int*)(dinv + N_NODES);          // [N]
  float* gsum   = (float*)(winner + N_NODES);      // [64]
  float* gsumsq = gsum + HID;                      // [64]

  const int* src = ei;             // edge_index[0]
  const int* dst = ei + N_EDGES;   // edge_index[1]

  const int nodeBlocks = (N_NODES + 255) / 256;
  const long ncElems = (long)N_NODES * HID;
  const int ncBlocks = (int)((ncElems + 255) / 256);
  const long ecElems = (long)N_EDGES * HID;
  const int ecBlocks = (int)((ecElems + 255) / 256);
  const int gemmBlocks = N_NODES / 16;  // 6250

  init_kernel<<<nodeBlocks, 256, 0, stream>>>(dinv, winner);
  edge_prep_kernel<<<(N_EDGES + 255) / 256, 256, 0, stream>>>(src, dst, ea, dinv, winner);
  gemm_n64_kernel<<<gemmBlocks, 128, 0, stream>>>(x, linW, linb, h);
  dinv_kernel<<<nodeBlocks, 256, 0, stream>>>(dinv);
  apply_edge_kernel<<<ncBlocks, 256, 0, stream>>>(h, winner, ea, ew, eb);

  for (int l = 0; l < 2; ++l) {
    gemm_n64_kernel<<<gemmBlocks, 128, 0, stream>>>(h, convW + (long)l * HID * HID, nullptr, xw);
    agg_init_kernel<<<ncBlocks, 256, 0, stream>>>(xw, dinv, convb + l * HID, agg);
    scatter_kernel<<<ecBlocks, 256, 0, stream>>>(xw, dinv, src, dst, ea, agg);
    zero_sums_kernel<<<1, 128, 0, stream>>>(gsum, gsumsq);
    bn_stats_kernel<<<1024, 256, 0, stream>>>(agg, gsum, gsumsq);
    bn_norm_kernel<<<ncBlocks, 256, 0, stream>>>(agg, gsum, gsumsq, gamma + l * HID,
                                                 beta + l * HID, h, out, l);
  }
}